// ContrastiveEmbeddingLoss_81947976007852
// MI455X (gfx1250) — compile-verified
//
#include <hip/hip_runtime.h>
#include <math.h>
#include <stdint.h>

typedef __attribute__((ext_vector_type(2))) float v2f;
typedef __attribute__((ext_vector_type(4))) float v4f;
typedef __attribute__((ext_vector_type(8))) float v8f;

#define N_PTS   8192
#define DIMS    128
#define INV_T   2.0f           // 1 / TEMPERATURE
#define EPSV    1e-8f
#define NCTILE  (N_PTS / 16)   // 512 column tiles
#define BSTRIDE 132            // padded LDS row stride (floats) -> conflict-free ds_load_b64

__global__ void cel_init_kernel(float* accum) {
    accum[0] = 0.0f;   // sum of valid losses
    accum[1] = 0.0f;   // valid count
}

// Issue this thread's share (4 x 16B) of one 16x128 B tile as async LDS loads.
__device__ __forceinline__ void stage_tile_async(const float* __restrict__ emb,
                                                 int ct, uint32_t lds_base_bytes, int tid) {
#pragma unroll
    for (int q = 0; q < 4; ++q) {
        int idx4 = tid + 128 * q;          // 0..511 float4 chunks
        int r    = idx4 >> 5;              // tile row 0..15
        int c4   = (idx4 & 31) * 4;        // dim offset 0..124
        uint64_t gaddr = (uint64_t)(uintptr_t)(emb + (size_t)(ct * 16 + r) * DIMS + c4);
        uint32_t laddr = lds_base_bytes + (uint32_t)((r * BSTRIDE + c4) * 4);
        asm volatile("global_load_async_to_lds_b128 %0, %1, off"
                     :: "v"(laddr), "v"(gaddr) : "memory");
    }
}

__device__ __forceinline__ void wait_async0() {
    asm volatile("s_wait_asynccnt 0x0" ::: "memory");
}

__global__ __launch_bounds__(128) void cel_main_kernel(const int* __restrict__ labels,
                                                       const float* __restrict__ emb,
                                                       float* __restrict__ accum) {
    __shared__ int s_lab[N_PTS];                                 // 32 KB label cache
    __shared__ __align__(16) float s_B[2][16 * BSTRIDE];         // double-buffered B tile

    const int tid  = threadIdx.x;
    const int lane = tid & 31;
    const int wave = tid >> 5;
    const int n16  = lane & 15;      // M index (A) / N index (B,C)
    const int half = lane >> 4;      // K-half selector

    const uint32_t ldsb0 = (uint32_t)(uintptr_t)(&s_B[0][0]);
    const uint32_t ldsb1 = (uint32_t)(uintptr_t)(&s_B[1][0]);

    // cache all labels in LDS
    for (int i = tid; i < N_PTS; i += 128) s_lab[i] = labels[i];

    const int rowbase = (blockIdx.x * 4 + wave) * 16;

    // ---- load A fragments for this wave's 16 rows, pre-scaled by 1/T ----
    // ISA 32-bit A 16x4 layout: lane = M (mod 16), VGPR v holds K = 4*s + 2*half + v
    v2f A[32];
    {
        const float* aptr = emb + (size_t)(rowbase + n16) * DIMS + 2 * half;
#pragma unroll
        for (int s = 0; s < 32; ++s) {
            v2f t = *(const v2f*)(aptr + 4 * s);
            A[s] = t * INV_T;
        }
    }

    // anchor labels for this lane's 8 C-slots (rows M = v + 8*half)
    int alab[8];
#pragma unroll
    for (int v = 0; v < 8; ++v) alab[v] = labels[rowbase + v + 8 * half];

    // online-softmax state per C-slot (this lane's column subset)
    float m[8], ps[8], ns[8], cnt[8];
#pragma unroll
    for (int v = 0; v < 8; ++v) { m[v] = -INFINITY; ps[v] = 0.f; ns[v] = 0.f; cnt[v] = 0.f; }

    // ---- prologue: stage tile 0, sync, then keep tile 1 in flight ----
    stage_tile_async(emb, 0, ldsb0, tid);
    wait_async0();
    __syncthreads();                 // tile 0 + labels visible to all waves
    stage_tile_async(emb, 1, ldsb1, tid);

    int cur = 0;
    for (int ct = 0; ct < NCTILE; ++ct) {
        // ---- 16x16 sim tile: two independent fp32 WMMA chains (K split) ----
        v8f acc0 = {}, acc1 = {};
        const float* bptr = &s_B[cur][n16 * BSTRIDE + 2 * half];
#pragma unroll
        for (int s = 0; s < 16; ++s) {
            v2f b0 = *(const v2f*)(bptr + 8 * s);
            v2f b1 = *(const v2f*)(bptr + 8 * s + 4);
            acc0 = __builtin_amdgcn_wmma_f32_16x16x4_f32(
                false, A[2 * s],     false, b0, (short)0, acc0, false, false);
            acc1 = __builtin_amdgcn_wmma_f32_16x16x4_f32(
                false, A[2 * s + 1], false, b1, (short)0, acc1, false, false);
        }
        v8f acc = acc0 + acc1;

        // ---- online softmax + pos/neg masking ----
        const int j  = ct * 16 + n16;      // this lane's global column
        const int tl = s_lab[j];
#pragma unroll
        for (int v = 0; v < 8; ++v) {
            float sim  = acc[v];                   // already includes 1/T
            int   gr   = rowbase + v + 8 * half;   // global row
            bool  diag = (gr == j);
            int   a    = alab[v];
            bool  ispos = (a == 0) || (tl == a) || (tl == 0);

            float newm = fmaxf(m[v], sim);         // max includes diagonal (matches ref)
            float sc   = expf(m[v] - newm);
            float e    = expf(sim - newm);
            ps[v] *= sc;
            ns[v] *= sc;
            m[v]  = newm;
            if (!diag) {
                if (ispos) { ps[v] += e; cnt[v] += 1.0f; }
                else       { ns[v] += e; }
            }
        }

        // ---- rotate double buffer: wait next tile's async loads, sync, refill ----
        wait_async0();                 // loads for tile ct+1 (other buffer) complete
        __syncthreads();               // all waves done reading buffer `cur`
        if (ct + 2 < NCTILE)
            stage_tile_async(emb, ct + 2, (cur == 0) ? ldsb0 : ldsb1, tid);
        cur ^= 1;
    }

    // ---- merge state across the 16 lanes of each half (butterfly) ----
#pragma unroll
    for (int off = 1; off < 16; off <<= 1) {
#pragma unroll
        for (int v = 0; v < 8; ++v) {
            float mo = __shfl_xor(m[v],   off, 32);
            float po = __shfl_xor(ps[v],  off, 32);
            float no = __shfl_xor(ns[v],  off, 32);
            float co = __shfl_xor(cnt[v], off, 32);
            float M  = fmaxf(m[v], mo);
            ps[v]  = ps[v] * expf(m[v] - M) + po * expf(mo - M);
            ns[v]  = ns[v] * expf(m[v] - M) + no * expf(mo - M);
            cnt[v] += co;
            m[v]   = M;
        }
    }

    // lanes 0 (rows 0..7) and 16 (rows 8..15) hold full row results
    if (n16 == 0) {
        float lsum = 0.f, vc = 0.f;
#pragma unroll
        for (int v = 0; v < 8; ++v) {
            bool  valid = cnt[v] > 0.f;
            float loss  = -logf((ps[v] + EPSV) / (ps[v] + ns[v] + EPSV));
            if (valid) { lsum += loss; vc += 1.0f; }
        }
        atomicAdd(&accum[0], lsum);
        atomicAdd(&accum[1], vc);
    }
}

__global__ void cel_finalize_kernel(const float* __restrict__ accum, float* __restrict__ out) {
    float c = accum[1];
    out[0] = (c > 0.f) ? (accum[0] / fmaxf(c, 1.0f)) : 0.0f;
}

extern "C" void kernel_launch(void* const* d_in, const int* in_sizes, int n_in,
                              void* d_out, int out_size, void* d_ws, size_t ws_size,
                              hipStream_t stream) {
    const int*   labels = (const int*)d_in[0];
    const float* emb    = (const float*)d_in[1];
    float*       out    = (float*)d_out;
    float*       accum  = (float*)d_ws;   // 2 floats of scratch

    cel_init_kernel<<<1, 1, 0, stream>>>(accum);
    cel_main_kernel<<<N_PTS / 64, 128, 0, stream>>>(labels, emb, accum);
    cel_finalize_kernel<<<1, 1, 0, stream>>>(accum, out);
}